// cAttend_simple_40922448396535
// MI455X (gfx1250) — compile-verified
//
#include <hip/hip_runtime.h>
#include <hip/hip_bf16.h>

#define B_DIM 4
#define N_DIM 4096
#define E_DIM 256
#define D_DIM 64
#define SCALE_F 0.125f   // 64^-0.5

typedef __attribute__((ext_vector_type(2))) float v2f;
typedef __attribute__((ext_vector_type(8))) float v8f;

// Workspace layout (floats):
//   [0    , 1024) : u[4][256]     (atomically accumulated)
//   [1024 , 1028) : vsum[4]       (atomically accumulated)
//   [1028 , 2052) : w[4][256]
//   [2052 , 2056) : c[4]
#define WS_U     0
#define WS_VSUM  1024
#define WS_W     1028
#define WS_C     2052

__global__ void zero_ws_kernel(float* __restrict__ ws) {
    int i = blockIdx.x * blockDim.x + threadIdx.x;
    if (i < WS_W) ws[i] = 0.0f;   // zero u + vsum only; w/c fully overwritten
}

// Phase A: u[b,e] = sum_i val[b,i]*embed[pos[b,i],e] ; vsum[b] = sum_i val[b,i]
// 128 blocks per batch, 32 rows per block, 256 threads (thread t owns column e=t).
__global__ void phaseA_kernel(const float* __restrict__ val,
                              const int*   __restrict__ pos,
                              const float* __restrict__ emb,
                              float*       __restrict__ ws) {
    __shared__ float vs[32];
    __shared__ int   ps[32];
    const int b  = blockIdx.x >> 7;          // 128 blocks per batch
    const int i0 = (blockIdx.x & 127) * 32;
    const int t  = threadIdx.x;
    if (t < 32) {
        vs[t] = val[b * N_DIM + i0 + t];
        ps[t] = pos[b * N_DIM + i0 + t];
    }
    __syncthreads();
    float acc = 0.0f;
    #pragma unroll 4
    for (int ii = 0; ii < 32; ++ii) {
        acc = fmaf(vs[ii], emb[ps[ii] * E_DIM + t], acc);   // coalesced 1KB row
    }
    atomicAdd(&ws[WS_U + b * E_DIM + t], acc);
    if (t == 0) {
        float s = 0.0f;
        for (int ii = 0; ii < 32; ++ii) s += vs[ii];
        atomicAdd(&ws[WS_VSUM + b], s);
    }
}

// Phase B: S[64x16] = Wq[64x256] x U[256x16] via v_wmma_f32_16x16x4_f32
// (only first 4 N-columns valid; rest zero-padded). Then fold bias, compute
// w[b,e] = sum_d s[b,d]*Wk[d,e] and c[b] = s[b].bk.  One block, 4 waves.
__global__ void phaseB_kernel(const float* __restrict__ Wq,
                              const float* __restrict__ bq,
                              const float* __restrict__ Wk,
                              const float* __restrict__ bk,
                              float*       __restrict__ ws) {
    __shared__ float s_lds[B_DIM][D_DIM];
    const float* u    = ws + WS_U;
    const float* vsum = ws + WS_VSUM;
    float*       wv   = ws + WS_W;
    float*       cv   = ws + WS_C;

    const int tid   = threadIdx.x;
    const int wave  = tid >> 5;                 // M-tile index: rows d = wave*16..+15
    const int lane  = tid & 31;
    const int mrow  = wave * 16 + (lane & 15);  // A: lanes 0-15 & 16-31 both hold M=0..15
    const int khalf = (lane >> 4) * 2;          // A/B: upper lane half holds K=2,3
    const int ncol  = lane & 15;                // B/D column (batch index, <4 valid)
    const int ucol  = (ncol < B_DIM) ? ncol : 0;
    const float nmask = (ncol < B_DIM) ? 1.0f : 0.0f;

    v8f acc = {};
    for (int k = 0; k < E_DIM; k += 4) {
        v2f a, bm;
        a.x  = Wq[mrow * E_DIM + k + khalf];
        a.y  = Wq[mrow * E_DIM + k + khalf + 1];
        bm.x = u[ucol * E_DIM + k + khalf] * nmask;
        bm.y = u[ucol * E_DIM + k + khalf + 1] * nmask;
        acc = __builtin_amdgcn_wmma_f32_16x16x4_f32(
                  /*neg_a=*/false, a, /*neg_b=*/false, bm,
                  /*c_mod=*/(short)0, acc, /*reuse_a=*/false, /*reuse_b=*/false);
    }
    // D layout: VGPR r holds M = r + (lane>=16 ? 8 : 0), N = lane&15
    if (ncol < B_DIM) {
        float vsb = vsum[ncol];
        #pragma unroll
        for (int r = 0; r < 8; ++r) {
            int d = wave * 16 + r + ((lane >> 4) << 3);
            s_lds[ncol][d] = acc[r] + bq[d] * vsb;
        }
    }
    __syncthreads();
    for (int e = tid; e < E_DIM; e += 128) {
        for (int b = 0; b < B_DIM; ++b) {
            float aw = 0.0f;
            #pragma unroll 8
            for (int d = 0; d < D_DIM; ++d)
                aw = fmaf(s_lds[b][d], Wk[d * E_DIM + e], aw);
            wv[b * E_DIM + e] = aw;
        }
    }
    if (tid < B_DIM) {
        float ac = 0.0f;
        for (int d = 0; d < D_DIM; ++d) ac = fmaf(s_lds[tid][d], bk[d], ac);
        cv[tid] = ac;
    }
}

// Phase C: out[b,j] = val[b,j] * (1 + scale*( w[b].embed[pos[b,j]] + c[b] ))
// One wave32 per output j; 8 waves (8 j's) per 256-thread block.
__global__ void phaseC_kernel(const float* __restrict__ val,
                              const int*   __restrict__ pos,
                              const float* __restrict__ emb,
                              const float* __restrict__ ws,
                              float*       __restrict__ out) {
    __shared__ float wsh[E_DIM];
    __shared__ float csh;
    const int b     = blockIdx.x >> 9;            // 512 blocks per batch
    const int jbase = (blockIdx.x & 511) * 8;
    const int tid   = threadIdx.x;
    wsh[tid] = ws[WS_W + b * E_DIM + tid];
    if (tid == 0) csh = ws[WS_C + b];
    __syncthreads();

    const int wave = tid >> 5;
    const int lane = tid & 31;
    const int j    = jbase + wave;
    const int base = pos[b * N_DIM + j] * E_DIM;

    float dot = 0.0f;
    #pragma unroll
    for (int s = 0; s < 8; ++s) {                 // coalesced 128B per wave step
        int e = lane + 32 * s;
        dot = fmaf(wsh[e], emb[base + e], dot);
    }
    #pragma unroll
    for (int off = 16; off > 0; off >>= 1)
        dot += __shfl_xor(dot, off, 32);
    if (lane == 0) {
        float v = val[b * N_DIM + j];
        out[b * N_DIM + j] = v * (1.0f + SCALE_F * (dot + csh));
    }
}

extern "C" void kernel_launch(void* const* d_in, const int* in_sizes, int n_in,
                              void* d_out, int out_size, void* d_ws, size_t ws_size,
                              hipStream_t stream) {
    (void)in_sizes; (void)n_in; (void)out_size; (void)ws_size;
    // setup_inputs order: t, val, pos, embed, Wq, bq, Wk, bk
    const float* val = (const float*)d_in[1];
    const int*   pos = (const int*)  d_in[2];
    const float* emb = (const float*)d_in[3];
    const float* Wq  = (const float*)d_in[4];
    const float* bq  = (const float*)d_in[5];
    const float* Wk  = (const float*)d_in[6];
    const float* bk  = (const float*)d_in[7];
    float* out = (float*)d_out;
    float* ws  = (float*)d_ws;

    zero_ws_kernel<<<5, 256, 0, stream>>>(ws);
    phaseA_kernel<<<B_DIM * 128, 256, 0, stream>>>(val, pos, emb, ws);
    phaseB_kernel<<<1, 128, 0, stream>>>(Wq, bq, Wk, bk, ws);
    phaseC_kernel<<<B_DIM * 512, 256, 0, stream>>>(val, pos, emb, ws, out);
}